// Remover_57947698757992
// MI455X (gfx1250) — compile-verified
//
#include <hip/hip_runtime.h>

// Streaming elementwise kernel:
//   out[p, i] = (x[p, i] != x[p, 0]) ? 1.0f : 0.0f
// for 48 planes (16 batches x 3 channels) of 1024x1024 fp32.
//
// Memory-bound: ~402.7 MB total traffic -> ~17.3 us floor at 23.3 TB/s.
// Strategy: 128-bit global loads/stores with non-temporal hints (single-touch
// stream, working set > 192 MB L2), uniform corner load kept cacheable,
// gfx1250 global_prefetch_b8 to pull ahead within each thread's chunk.

typedef float v4f __attribute__((ext_vector_type(4)));

#define PLANE_ELEMS   (1024 * 1024)   // floats per (b,c) plane
#define THREADS       256             // 8 wave32s per block
#define ITERS         4               // float4s per thread
#define VEC4_PER_BLK  (THREADS * ITERS)   // 1024 float4 = 4096 floats / block

__global__ __launch_bounds__(THREADS) void
remover_plane_kernel(const float* __restrict__ x, float* __restrict__ out) {
    // Plane selection: blockIdx.y in [0, 48). Plane base is uniform per block.
    const size_t planeBase = (size_t)blockIdx.y * (size_t)PLANE_ELEMS;

    // Corner value: uniform address -> scalar/cached load; reused by all 256
    // blocks of this plane, so do NOT mark it non-temporal.
    const float corner = x[planeBase];

    const v4f* __restrict__ vin  = (const v4f*)(x + planeBase);
    v4f*       __restrict__ vout = (v4f*)(out + planeBase);

    // Each block covers VEC4_PER_BLK consecutive float4s of the plane,
    // striped across threads for fully coalesced b128 accesses.
    const int base = blockIdx.x * VEC4_PER_BLK + (int)threadIdx.x;

    // gfx1250 prefetch (lowers to global_prefetch_b8): pull the back half of
    // this thread's chunk toward the caches while the front half streams.
    __builtin_prefetch(&vin[base + 2 * THREADS], 0, 0);

#pragma unroll
    for (int i = 0; i < ITERS; ++i) {
        const int idx = base + i * THREADS;

        // Single-touch stream: non-temporal load + store (TH=NT) so the
        // 402 MB stream does not churn the 192 MB L2.
        v4f v = __builtin_nontemporal_load(&vin[idx]);

        v4f r;
        r.x = (v.x != corner) ? 1.0f : 0.0f;
        r.y = (v.y != corner) ? 1.0f : 0.0f;
        r.z = (v.z != corner) ? 1.0f : 0.0f;
        r.w = (v.w != corner) ? 1.0f : 0.0f;

        __builtin_nontemporal_store(r, &vout[idx]);
    }
}

extern "C" void kernel_launch(void* const* d_in, const int* in_sizes, int n_in,
                              void* d_out, int out_size, void* d_ws, size_t ws_size,
                              hipStream_t stream) {
    (void)n_in; (void)d_ws; (void)ws_size; (void)out_size;

    const float* x   = (const float*)d_in[0];
    float*       out = (float*)d_out;

    const int nPlanes      = in_sizes[0] / PLANE_ELEMS;                  // 48
    const int blksPerPlane = PLANE_ELEMS / (VEC4_PER_BLK * 4);           // 256

    dim3 grid(blksPerPlane, nPlanes);   // (256, 48) = 12288 blocks
    dim3 block(THREADS);                // 8 wave32s

    remover_plane_kernel<<<grid, block, 0, stream>>>(x, out);
}